// Interpolation_Layer_30124900614766
// MI455X (gfx1250) — compile-verified
//
#include <hip/hip_runtime.h>
#include <hip/hip_bf16.h>

#define NK    1024                 // knot count (fixed by reference)
#define BLOCK 256                  // 8 waves per workgroup on wave32
#define WS_FLOATS (NK + 4 * NK)    // search array + AoS quads = 5120 floats (20 KB)

// ---- CDNA5 async global->LDS staging (ASYNCcnt path), guarded -------------
#if defined(__has_builtin)
#if __has_builtin(__builtin_amdgcn_global_load_async_to_lds_b32) && \
    __has_builtin(__builtin_amdgcn_s_wait_asynccnt)
#define USE_ASYNC_LDS 1
#else
#define USE_ASYNC_LDS 0
#endif
#if __has_builtin(__builtin_amdgcn_global_load_async_to_lds_b128)
#define USE_ASYNC_B128 1
#else
#define USE_ASYNC_B128 0
#endif
#else
#define USE_ASYNC_LDS 0
#define USE_ASYNC_B128 0
#endif

typedef int v4i __attribute__((vector_size(16)));
typedef __attribute__((address_space(1))) int*  gptr_i32;   // "__device__ int *"
typedef __attribute__((address_space(3))) int*  lptr_i32;   // "__shared__ int *"
typedef __attribute__((address_space(1))) v4i*  gptr_v4i;   // "__device__ v4i *"
typedef __attribute__((address_space(3))) v4i*  lptr_v4i;   // "__shared__ v4i *"

// ---------------------------------------------------------------------------
// Prep kernel: one block of 1024 threads.
//  - rank-sort the 1024 knots (stable: ties broken by original index)
//  - ws layout (floats):
//      [0 .. NK)      : sorted x (binary-search array)
//      [NK .. 5*NK)   : per-interval quads (x1, y1, slope, 0), 16B stride;
//                       quad[NK-1] = (y_first, y_last, 0, 0)
// ---------------------------------------------------------------------------
__global__ __launch_bounds__(NK)
void interp_prep_kernel(const float* __restrict__ xp,
                        const float* __restrict__ yp,
                        float* __restrict__ ws) {
    __shared__ float xv[NK];
    __shared__ float yv[NK];
    __shared__ float sxs[NK];
    __shared__ float sys[NK];

    const int t = threadIdx.x;
    xv[t] = xp[t];
    yv[t] = yp[t];
    __syncthreads();

    const float myx = xv[t];
    int rank = 0;
    #pragma unroll 8
    for (int j = 0; j < NK; ++j) {
        const float xj = xv[j];
        rank += (xj < myx) || (xj == myx && j < t);
    }
    sxs[rank] = myx;
    sys[rank] = yv[t];
    __syncthreads();

    const float sx_t = sxs[t];
    const float sy_t = sys[t];
    ws[t] = sx_t;                      // search array

    if (t < NK - 1) {
        const float x2 = sxs[t + 1];
        const float y2 = sys[t + 1];
        const float m  = (y2 - sy_t) / (x2 - sx_t);
        ws[NK + 4 * t + 0] = sx_t;     // x1
        ws[NK + 4 * t + 1] = sy_t;     // y1
        ws[NK + 4 * t + 2] = m;        // slope
        ws[NK + 4 * t + 3] = 0.0f;
    } else {                           // quad NK-1 is never an interval: stash boundaries
        ws[NK + 4 * t + 0] = sys[0];   // y_first
        ws[NK + 4 * t + 1] = sy_t;     // y_last (t == NK-1, sorted)
        ws[NK + 4 * t + 2] = 0.0f;
        ws[NK + 4 * t + 3] = 0.0f;
    }
}

// ---------------------------------------------------------------------------
// Main streaming kernel: grid-stride over float4 query chunks.
// Each block stages the 20KB table into LDS (async-to-LDS on CDNA5),
// caches binary-search levels 1-3 in registers, then:
//   search = 7x ds_load_b32 + 1 final b32, epilogue = 1x ds_load_b128 + fma.
// ---------------------------------------------------------------------------
__global__ __launch_bounds__(BLOCK)
void interp_main_kernel(const float* __restrict__ x,
                        const float* __restrict__ tbl,   // ws: WS_FLOATS floats
                        float* __restrict__ out,
                        int nvec, int n) {
    __shared__ __align__(16) float tab[WS_FLOATS];   // 20 KB of 320 KB per WGP

    const int t = threadIdx.x;

#if USE_ASYNC_LDS && USE_ASYNC_B128
    #pragma unroll
    for (int k = 0; k < WS_FLOATS / (4 * BLOCK); ++k) {   // 5 x b128 per thread
        const int i = (k * BLOCK + t) * 4;
        __builtin_amdgcn_global_load_async_to_lds_b128(
            (gptr_v4i)(v4i*)(tbl + i), (lptr_v4i)(v4i*)(&tab[i]),
            /*offset=*/0, /*cpol=*/0);
    }
    __builtin_amdgcn_s_wait_asynccnt(0);
    __syncthreads();
#elif USE_ASYNC_LDS
    #pragma unroll
    for (int k = 0; k < WS_FLOATS / BLOCK; ++k) {         // 20 x b32 per thread
        const int i = k * BLOCK + t;
        __builtin_amdgcn_global_load_async_to_lds_b32(
            (gptr_i32)(int*)(tbl + i), (lptr_i32)(int*)(&tab[i]),
            /*offset=*/0, /*cpol=*/0);
    }
    __builtin_amdgcn_s_wait_asynccnt(0);
    __syncthreads();
#else
    #pragma unroll
    for (int k = 0; k < WS_FLOATS / BLOCK; ++k) {
        const int i = k * BLOCK + t;
        tab[i] = tbl[i];
    }
    __syncthreads();
#endif

    const float4* __restrict__ aos = (const float4*)(tab + NK);  // byte 4096: 16B aligned

    // boundary values + register-resident top 3 search levels (LDS broadcast reads)
    const float y_first = aos[NK - 1].x;
    const float y_last  = aos[NK - 1].y;
    const float c511 = tab[511];
    const float c255 = tab[255], c767 = tab[767];
    const float c127 = tab[127], c383 = tab[383], c639 = tab[639], c895 = tab[895];

    auto interp_one = [&](float xq) -> float {
        // levels 1-3 from registers
        const bool b1 = (c511 < xq);
        int base = b1 ? 512 : 0;
        const float l2 = b1 ? c767 : c255;
        const bool b2 = (l2 < xq);
        base += b2 ? 256 : 0;
        const float l3 = b2 ? (b1 ? c895 : c383) : (b1 ? c639 : c127);
        base += (l3 < xq) ? 128 : 0;
        // levels 4-10 from LDS
        #pragma unroll
        for (int step = 64; step >= 1; step >>= 1) {
            base += (tab[base + step - 1] < xq) ? step : 0;
        }
        const int idx = base + ((tab[base] < xq) ? 1 : 0);  // searchsorted(..., 'left')

        int hi = idx;
        if (hi < 1)      hi = 1;
        if (hi > NK - 1) hi = NK - 1;
        const float4 c = aos[hi - 1];                       // (x1, y1, m, -)
        float r = fmaf(xq - c.x, c.z, c.y);                 // exact xq-x1 (Sterbenz)

        r = (idx == 0)  ? y_first : r;
        r = (idx >= NK) ? y_last  : r;
        return r;
    };

    const int gid    = blockIdx.x * BLOCK + t;
    const int stride = gridDim.x * BLOCK;

    const float4* __restrict__ x4 = (const float4*)x;
    float4* __restrict__ o4 = (float4*)out;

    for (int i = gid; i < nvec; i += stride) {
        // prefetch next chunk for this thread (global_prefetch_b8 on gfx1250)
        __builtin_prefetch(&x4[i + stride], 0, 1);

        const float4 q = x4[i];
        float4 r;
        r.x = interp_one(q.x);
        r.y = interp_one(q.y);
        r.z = interp_one(q.z);
        r.w = interp_one(q.w);
        o4[i] = r;
    }

    // scalar tail (empty for n = 32M, kept for generality)
    for (int i = 4 * nvec + gid; i < n; i += stride) {
        out[i] = interp_one(x[i]);
    }
}

// ---------------------------------------------------------------------------
extern "C" void kernel_launch(void* const* d_in, const int* in_sizes, int n_in,
                              void* d_out, int out_size, void* d_ws, size_t ws_size,
                              hipStream_t stream) {
    const float* x  = (const float*)d_in[0];
    const float* xp = (const float*)d_in[1];
    const float* yp = (const float*)d_in[2];
    float* out = (float*)d_out;
    float* ws  = (float*)d_ws;
    const int n = in_sizes[0];

    // 1) sort knots + build search array / interval quads in workspace
    interp_prep_kernel<<<1, NK, 0, stream>>>(xp, yp, ws);

    // 2) stream the queries
    const int nvec = n >> 2;
    int grid = (nvec + BLOCK - 1) / BLOCK;
    if (grid > 8192) grid = 8192;   // grid-stride: amortize LDS table staging
    if (grid < 1)    grid = 1;
    interp_main_kernel<<<grid, BLOCK, 0, stream>>>(x, ws, out, nvec, n);
}